// MK_our_union_86483461472748
// MI455X (gfx1250) — compile-verified
//
#include <hip/hip_runtime.h>
#include <hip/hip_bf16.h>

typedef float v2f __attribute__((ext_vector_type(2)));
typedef float v8f __attribute__((ext_vector_type(8)));

#define NWORDS   (1u << 20)   // 2^25 ids / 32 bits per word
#define NGROUPS  1024         // NWORDS / WPG
#define WPG      1024u        // bitmap words per group
#define THREADS  256
#define STAGE_LDS_BYTES ((2048 + 256) * 4)   // 2x1024 staged words + 256 scan slots

// ---------------------------------------------------------------------------
// Pass 0: zero the two presence bitmaps (they are contiguous in workspace)
// ---------------------------------------------------------------------------
__global__ void MK_k_zero(unsigned* __restrict__ p, int n) {
    int i = blockIdx.x * blockDim.x + threadIdx.x;
    if (i < n) p[i] = 0u;
}

// ---------------------------------------------------------------------------
// Pass 1: build per-set presence bitmaps from the coordinate rows
// id = ((b*256 + x)*256 + y)*256 + z  (monotone reordering of the ref hash)
// ---------------------------------------------------------------------------
__global__ void MK_k_build_bitmaps(const int* __restrict__ C1, const int* __restrict__ C2,
                                   int n1, int n2,
                                   unsigned* __restrict__ bm1, unsigned* __restrict__ bm2) {
    int i = blockIdx.x * blockDim.x + threadIdx.x;
    if (i < n1) {
        const int* c = C1 + 4 * (size_t)i;
        unsigned id = (((unsigned)c[0] * 256u + (unsigned)c[1]) * 256u + (unsigned)c[2]) * 256u
                      + (unsigned)c[3];
        atomicOr(&bm1[id >> 5], 1u << (id & 31u));
    } else if (i < n1 + n2) {
        const int* c = C2 + 4 * (size_t)(i - n1);
        unsigned id = (((unsigned)c[0] * 256u + (unsigned)c[1]) * 256u + (unsigned)c[2]) * 256u
                      + (unsigned)c[3];
        atomicOr(&bm2[id >> 5], 1u << (id & 31u));
    }
}

// ---------------------------------------------------------------------------
// Async-stage 16 bytes/lane of each bitmap into LDS (ASYNCcnt path).
// Dynamic LDS only => region starts at LDS byte offset 0, so the per-lane
// LDS destination address is a plain integer (no addrspace casts needed).
// ---------------------------------------------------------------------------
__device__ __forceinline__ void MK_stage_bitmaps_async(const unsigned* __restrict__ bm1,
                                                       const unsigned* __restrict__ bm2,
                                                       unsigned gword) {
    unsigned t = threadIdx.x;
    unsigned lds0 = t * 16u;           // staging region A: bytes [0, 4096)
    unsigned lds1 = 4096u + t * 16u;   // staging region B: bytes [4096, 8192)
    unsigned long long ga1 = (unsigned long long)(const void*)(bm1 + gword);
    unsigned long long ga2 = (unsigned long long)(const void*)(bm2 + gword);
    asm volatile("global_load_async_to_lds_b128 %0, %1, off"
                 :: "v"(lds0), "v"(ga1) : "memory");
    asm volatile("global_load_async_to_lds_b128 %0, %1, off"
                 :: "v"(lds1), "v"(ga2) : "memory");
    asm volatile("s_wait_asynccnt 0x0" ::: "memory");
    __syncthreads();
}

// ---------------------------------------------------------------------------
// Pass 2: per-group (1024 words) popcount of union bitmap -> groupSum (f32)
// ---------------------------------------------------------------------------
__global__ void MK_k_group_reduce(const unsigned* __restrict__ bm1,
                                  const unsigned* __restrict__ bm2,
                                  float* __restrict__ groupSum) {
    extern __shared__ unsigned smem[];              // [0,1024) bm1, [1024,2048) bm2, [2048,2304) scan
    unsigned g = blockIdx.x, t = threadIdx.x;
    unsigned base = g * WPG + t * 4u;
    MK_stage_bitmaps_async(bm1, bm2, base);
    uint4 a = *(const uint4*)&smem[t * 4u];
    uint4 b = *(const uint4*)&smem[1024u + t * 4u];
    unsigned s = (unsigned)__popc(a.x | b.x) + (unsigned)__popc(a.y | b.y)
               + (unsigned)__popc(a.z | b.z) + (unsigned)__popc(a.w | b.w);
    unsigned* r = smem + 2048;
    r[t] = s;
    __syncthreads();
    for (unsigned off = THREADS / 2; off > 0; off >>= 1) {
        if (t < off) r[t] += r[t + off];
        __syncthreads();
    }
    if (t == 0) groupSum[g] = (float)r[0];
}

// ---------------------------------------------------------------------------
// Pass 3: exclusive scan of 1024 group sums via matrix core.
// Inclusive row-scan of each 16x16 tile as Y = X * U (U upper-triangular
// ones), built from 4 chained V_WMMA_F32_16X16X4_F32 ops; then a tiny LDS
// fixup carries row/tile offsets.  Counts <= 4M are exact in f32.
// Single wave (32 lanes), EXEC all ones at every WMMA.
// ---------------------------------------------------------------------------
__global__ void MK_k_wmma_scan(const float* __restrict__ groupSum,
                               float* __restrict__ groupOff) {
    __shared__ float sX[256];
    __shared__ float sTot[16];
    __shared__ float sOff[16];
    __shared__ float sCarry;
    const unsigned lane = threadIdx.x;   // 0..31
    const unsigned g    = lane >> 4;     // half-wave group
    const unsigned n    = lane & 15u;    // column index (also A-row index)
    if (lane == 0) sCarry = 0.0f;
    for (int tile = 0; tile < NGROUPS / 256; ++tile) {
        const int tb = tile * 256;
        // stage tile: X(m, c) = groupSum[tb + m*16 + c]
        #pragma unroll
        for (int j = 0; j < 8; ++j) sX[lane * 8 + j] = groupSum[tb + (int)lane * 8 + j];
        __syncthreads();

        v8f c = {};
        #pragma unroll
        for (int kt = 0; kt < 4; ++kt) {
            const int kb = kt * 4 + (int)(g * 2u);   // this lane's two K indices: kb, kb+1
            v2f a, b;
            a.x = sX[(int)n * 16 + kb + 0];          // A(m = lane&15, K)
            a.y = sX[(int)n * 16 + kb + 1];
            b.x = (kb + 0 <= (int)n) ? 1.0f : 0.0f;  // U(K, n) upper-triangular ones
            b.y = (kb + 1 <= (int)n) ? 1.0f : 0.0f;
            c = __builtin_amdgcn_wmma_f32_16x16x4_f32(
                    /*neg_a=*/false, a, /*neg_b=*/false, b,
                    /*c_mod=*/(short)0, c, /*reuse_a=*/false, /*reuse_b=*/false);
        }
        // c[v] = inclusive row scan Y(m = v + 8g, n).  Row totals live at n==15.
        if (n == 15u) {
            #pragma unroll
            for (int v = 0; v < 8; ++v) sTot[v + 8 * (int)g] = c[v];
        }
        __syncthreads();
        if (lane == 0) {
            float run = sCarry;
            for (int rr = 0; rr < 16; ++rr) { float tv = sTot[rr]; sOff[rr] = run; run += tv; }
            sCarry = run;
        }
        __syncthreads();
        // exclusive global value = inclusive_row - x + row_offset(incl. carry)
        #pragma unroll
        for (int v = 0; v < 8; ++v) {
            const int m = v + 8 * (int)g;
            groupOff[tb + m * 16 + (int)n] = c[v] - sX[m * 16 + (int)n] + sOff[m];
        }
        __syncthreads();
    }
}

// ---------------------------------------------------------------------------
// Pass 4: per-word exclusive offsets = groupOff[g] + intra-group LDS scan
// ---------------------------------------------------------------------------
__global__ void MK_k_word_offsets(const unsigned* __restrict__ bm1,
                                  const unsigned* __restrict__ bm2,
                                  const float* __restrict__ groupOff,
                                  unsigned* __restrict__ wordOff) {
    extern __shared__ unsigned smem[];              // same layout as group_reduce
    unsigned g = blockIdx.x, t = threadIdx.x;
    unsigned base = g * WPG + t * 4u;
    MK_stage_bitmaps_async(bm1, bm2, base);
    uint4 a = *(const uint4*)&smem[t * 4u];
    uint4 b = *(const uint4*)&smem[1024u + t * 4u];
    unsigned pc0 = (unsigned)__popc(a.x | b.x);
    unsigned pc1 = (unsigned)__popc(a.y | b.y);
    unsigned pc2 = (unsigned)__popc(a.z | b.z);
    unsigned pc3 = (unsigned)__popc(a.w | b.w);
    unsigned mySum = pc0 + pc1 + pc2 + pc3;
    unsigned* s = smem + 2048;
    s[t] = mySum;
    __syncthreads();
    for (unsigned off = 1; off < THREADS; off <<= 1) {   // Hillis-Steele inclusive
        unsigned x = (t >= off) ? s[t - off] : 0u;
        __syncthreads();
        s[t] += x;
        __syncthreads();
    }
    unsigned o = (unsigned)groupOff[g] + (s[t] - mySum);
    wordOff[base + 0] = o; o += pc0;
    wordOff[base + 1] = o; o += pc1;
    wordOff[base + 2] = o; o += pc2;
    wordOff[base + 3] = o;
}

// ---------------------------------------------------------------------------
// Pass 5: rank each row via bitmap popcount + scatter coords/features to the
// sorted position; masks = bit test in the *other* set's bitmap.
// ---------------------------------------------------------------------------
__global__ void MK_k_scatter(const int* __restrict__ C1, const float* __restrict__ F1,
                             const int* __restrict__ C2, const float* __restrict__ F2,
                             int n1, int n2,
                             const unsigned* __restrict__ bm1, const unsigned* __restrict__ bm2,
                             const unsigned* __restrict__ wordOff,
                             float* __restrict__ outC, float* __restrict__ outF,
                             float* __restrict__ mask1, float* __restrict__ mask2) {
    int i = blockIdx.x * blockDim.x + threadIdx.x;
    int n = n1 + n2;
    if (i >= n) return;
    const int* C; const float* F; int li;
    if (i < n1) { C = C1; F = F1; li = i; } else { C = C2; F = F2; li = i - n1; }

    __builtin_prefetch(F + (size_t)li * 8, 0, 0);   // global_prefetch_b8 on feature row

    const int* cr = C + 4 * (size_t)li;
    int c0 = cr[0], c1 = cr[1], c2 = cr[2], c3 = cr[3];
    unsigned id = (((unsigned)c0 * 256u + (unsigned)c1) * 256u + (unsigned)c2) * 256u
                  + (unsigned)c3;
    unsigned w = id >> 5, bit = id & 31u;
    unsigned w1 = bm1[w], w2 = bm2[w];
    unsigned un = w1 | w2;
    unsigned rank = wordOff[w] + (unsigned)__popc(un & ((1u << bit) - 1u));

    float4 cf = make_float4((float)c0, (float)c1, (float)c2, (float)c3);
    *(float4*)(outC + (size_t)rank * 4) = cf;
    float4 f0 = *(const float4*)(F + (size_t)li * 8);
    float4 f1 = *(const float4*)(F + (size_t)li * 8 + 4);
    *(float4*)(outF + (size_t)rank * 8)     = f0;
    *(float4*)(outF + (size_t)rank * 8 + 4) = f1;

    if (i < n1) mask1[i]  = ((w2 >> bit) & 1u) ? 1.0f : 0.0f;
    else        mask2[li] = ((w1 >> bit) & 1u) ? 1.0f : 0.0f;
}

// ---------------------------------------------------------------------------
extern "C" void kernel_launch(void* const* d_in, const int* in_sizes, int n_in,
                              void* d_out, int out_size, void* d_ws, size_t ws_size,
                              hipStream_t stream) {
    const int*   C1 = (const int*)d_in[0];
    const float* F1 = (const float*)d_in[1];
    const int*   C2 = (const int*)d_in[2];
    const float* F2 = (const float*)d_in[3];
    const int n1 = in_sizes[0] / 4;
    const int n2 = in_sizes[2] / 4;
    const int n  = n1 + n2;

    unsigned char* ws = (unsigned char*)d_ws;
    unsigned* bm1      = (unsigned*)(ws);                          // 4 MB
    unsigned* bm2      = (unsigned*)(ws + (size_t)NWORDS * 4);     // 4 MB
    unsigned* wordOff  = (unsigned*)(ws + (size_t)NWORDS * 8);     // 4 MB
    float*    groupSum = (float*)   (ws + (size_t)NWORDS * 12);    // 4 KB
    float*    groupOff = groupSum + NGROUPS;                       // 4 KB

    float* outC  = (float*)d_out;                 // [n,4]
    float* outF  = outC + (size_t)n * 4;          // [n,8]
    float* mask1 = outF + (size_t)n * 8;          // [n1]
    float* mask2 = mask1 + n1;                    // [n2]

    const int zeroN = (int)(2u * NWORDS);
    MK_k_zero<<<(zeroN + THREADS - 1) / THREADS, THREADS, 0, stream>>>(bm1, zeroN);
    MK_k_build_bitmaps<<<(n + THREADS - 1) / THREADS, THREADS, 0, stream>>>(
        C1, C2, n1, n2, bm1, bm2);
    MK_k_group_reduce<<<NGROUPS, THREADS, STAGE_LDS_BYTES, stream>>>(bm1, bm2, groupSum);
    MK_k_wmma_scan<<<1, 32, 0, stream>>>(groupSum, groupOff);
    MK_k_word_offsets<<<NGROUPS, THREADS, STAGE_LDS_BYTES, stream>>>(
        bm1, bm2, groupOff, wordOff);
    MK_k_scatter<<<(n + THREADS - 1) / THREADS, THREADS, 0, stream>>>(
        C1, F1, C2, F2, n1, n2, bm1, bm2, wordOff, outC, outF, mask1, mask2);
}